// Align_pool_mutil_avr_18631568130147
// MI455X (gfx1250) — compile-verified
//
#include <hip/hip_runtime.h>
#include <hip/hip_bf16.h>

// ---------------------------------------------------------------------------
// Fused ROIAlign-1D (multi-scale) + per-scale projection for MI455X (gfx1250).
//
//   out[n, j*128 + o] = b_j[o] + sum_{i,t} pooled_j[n,i,t] * w_j[o,i,t]
//
// One launch per scale. Each block: 16 ROIs.
//   Phase 0: precompute interpolation taps (low idx, wLo, wHi) in LDS.
//   Phase 1: gather/pool x (L2-resident, 128MB < 192MB L2) into an LDS tile
//            laid out K-major: pooled[k][roi], k = i*s + t  (B-matrix order).
//   Phase 2: 8 waves; each wave computes one 16(o) x 16(n) tile with
//            v_wmma_f32_16x16x4_f32 (exact fp32 — GEMM is only 2.7 GFLOP,
//            the gather dominates, so no precision downgrade needed).
// ---------------------------------------------------------------------------

typedef float v2f __attribute__((ext_vector_type(2)));
typedef float v8f __attribute__((ext_vector_type(8)));

#define T_LEN   4096
#define C_TOT   512
#define CGRP    128
#define NSAMP   4

__global__ __launch_bounds__(256)
void align_pool_wmma_kernel(const float* __restrict__ x,
                            const float* __restrict__ rois,
                            const float* __restrict__ w,      // (128, 128, s) row-major
                            const float* __restrict__ bias,   // (128,)
                            float* __restrict__ out,          // (N, 512)
                            int s, int j, int N)
{
    extern __shared__ float lds[];
    const int K = CGRP * s;                       // GEMM reduction length
    float* pooled = lds;                          // K * 16 floats (k-major, 16 ROIs)
    int*   lowIdx = (int*)  (lds + (size_t)K * 16);      // 16*s*4
    float* wLo    = (float*)(lowIdx + 64 * s);           // 16*s*4
    float* wHi    = wLo + 64 * s;                        // 16*s*4
    int*   roiB   = (int*)  (wHi + 64 * s);              // 16

    const int tid = threadIdx.x;
    const int n0  = blockIdx.x * 16;

    // ---- Phase 0: per-(roi, bin, sample) interpolation taps ----------------
    const int nItems = 16 * s * NSAMP;
    for (int item = tid; item < nItems; item += blockDim.x) {
        int roi = item / (s * NSAMP);
        int rem = item - roi * s * NSAMP;
        int p   = rem / NSAMP;
        int smp = rem - p * NSAMP;
        int rn  = n0 + roi; if (rn >= N) rn = N - 1;

        float bf    = rois[rn * 3 + 0];
        float start = rois[rn * 3 + 1];
        float end   = rois[rn * 3 + 2] - 1.0f;          // reference: end -= 1
        if (p == 0 && smp == 0) roiB[roi] = (int)bf;

        float length = fmaxf(end - start, 1.0f);
        float bin    = length / (float)s;
        float pos    = start + bin * ((float)p + ((float)smp + 0.5f) * (1.0f / NSAMP));
        bool  valid  = (pos >= -1.0f) && (pos <= (float)T_LEN);
        float pc     = fminf(fmaxf(pos, 0.0f), (float)(T_LEN - 1));
        int   low    = (int)floorf(pc);
        float wh     = pc - (float)low;
        float wl     = 1.0f - wh;
        if (!valid) { wl = 0.0f; wh = 0.0f; }
        lowIdx[item] = low;
        wLo[item]    = wl;
        wHi[item]    = wh;
    }
    __syncthreads();

    // ---- Phase 1: pool channels into LDS tile (B-matrix order) -------------
    {
        const int ch   = tid & (CGRP - 1);
        const int half = tid >> 7;                    // two (roi,t) pairs in flight
        const int P    = 16 * s;
        for (int pair = half; pair < P; pair += 2) {
            int roi = pair / s;
            int t   = pair - roi * s;
            const float* xr =
                x + ((size_t)(roiB[roi] * C_TOT + j * CGRP + ch)) * T_LEN;
            float acc = 0.0f;
            #pragma unroll
            for (int smp = 0; smp < NSAMP; ++smp) {
                int item = pair * NSAMP + smp;
                int lo   = lowIdx[item];
                int hi   = lo + 1; if (hi > T_LEN - 1) hi = T_LEN - 1;
                acc += wLo[item] * xr[lo] + wHi[item] * xr[hi];
            }
            pooled[(ch * s + t) * 16 + roi] = acc * (1.0f / NSAMP);
        }
    }
    __syncthreads();

    // ---- Phase 2: GEMM via V_WMMA_F32_16X16X4_F32 --------------------------
    // Wave wv computes D(16x16): rows o = wv*16 .. wv*16+15, cols n = n0..n0+15.
    // A (16x4, f32): lanes 0-15 hold K={k,k+1}, lanes 16-31 hold K={k+2,k+3}.
    // B (4x16, f32): VGPR0 = rows {k, k+2}, VGPR1 = rows {k+1, k+3} (half-lane split).
    const int lane  = tid & 31;
    const int wv    = tid >> 5;             // 0..7
    const int oBase = wv * 16;
    const int l16   = lane & 15;
    const int hi2   = (lane >> 4) * 2;      // 0 or 2

    const float* wRow = w + (size_t)(oBase + l16) * K + hi2;

    v8f c = {0.f, 0.f, 0.f, 0.f, 0.f, 0.f, 0.f, 0.f};
    #pragma unroll 2
    for (int k = 0; k < K; k += 4) {
        v2f a = *(const v2f*)(wRow + k);                 // global_load_b64, contiguous
        v2f b;
        b.x = pooled[(k + hi2 + 0) * 16 + l16];          // ds_load, conflict-free
        b.y = pooled[(k + hi2 + 1) * 16 + l16];
        c = __builtin_amdgcn_wmma_f32_16x16x4_f32(
                /*neg_a=*/false, a, /*neg_b=*/false, b,
                /*c_mod=*/(short)0, c, /*reuse_a=*/false, /*reuse_b=*/false);
    }

    // D layout: VGPR v -> o = oBase + v + 8*(lane>=16), n = n0 + l16.
    const int n = n0 + l16;
    if (n < N) {
        float*       orow = out  + (size_t)n * C_TOT + j * CGRP + oBase + (hi2 ? 8 : 0);
        const float* bp   = bias + oBase + (hi2 ? 8 : 0);
        #pragma unroll
        for (int v = 0; v < 8; ++v)
            orow[v] = c[v] + bp[v];
    }
}

extern "C" void kernel_launch(void* const* d_in, const int* in_sizes, int n_in,
                              void* d_out, int out_size, void* d_ws, size_t ws_size,
                              hipStream_t stream) {
    const float* x    = (const float*)d_in[0];
    const float* rois = (const float*)d_in[1];
    const float* ws4[4] = { (const float*)d_in[2], (const float*)d_in[4],
                            (const float*)d_in[6], (const float*)d_in[8] };
    const float* bs4[4] = { (const float*)d_in[3], (const float*)d_in[5],
                            (const float*)d_in[7], (const float*)d_in[9] };
    float* out = (float*)d_out;

    const int N = in_sizes[1] / 3;
    const int scales[4] = {1, 3, 7, 9};
    const int nTiles = (N + 15) / 16;

    for (int j = 0; j < 4; ++j) {
        const int s = scales[j];
        // pooled: 2048*s floats, taps: 3*64*s words, roiB: 16 words
        const size_t ldsBytes = (size_t)4 * (2048u * s + 192u * s + 16u);
        align_pool_wmma_kernel<<<dim3(nTiles), dim3(256), ldsBytes, stream>>>(
            x, rois, ws4[j], bs4[j], out, s, j, N);
    }
}